// BiDAF_75771813036897
// MI455X (gfx1250) — compile-verified
//
#include <hip/hip_runtime.h>

typedef float v2f __attribute__((ext_vector_type(2)));
typedef float v8f __attribute__((ext_vector_type(8)));

#define B_   128
#define LP_  400
#define LQ_  100
#define H2_  256

__device__ __forceinline__ v8f wmma_f32(v2f a, v2f b, v8f c) {
  // V_WMMA_F32_16X16X4_F32: D = A(16x4 f32) * B(4x16 f32) + C(16x16 f32)
  return __builtin_amdgcn_wmma_f32_16x16x4_f32(
      /*neg_a=*/false, a, /*neg_b=*/false, b,
      /*c_mod=*/(short)0, c, /*reuse_a=*/false, /*reuse_b=*/false);
}

// ---------------------------------------------------------------------------
// k1: qwT[b, d, lq] = sum_h w[h, d] * q[b, lq, h]
//     GEMM: A[d][h] = w[h][d] (M=256), B[h][lq] = q[b,lq,h] (N=100, K=256)
//     tiles per batch: 16 (d) x 7 (lq) = 112 ; 8 waves/block -> 14 blocks/batch
//     Invalid N columns (lq >= 100) load from a clamped address (no EXEC
//     divergence in the hot loop); their results are simply never stored.
// ---------------------------------------------------------------------------
__global__ __launch_bounds__(256) void k1_qwT(const float* __restrict__ q,
                                              const float* __restrict__ w,
                                              float* __restrict__ qwT) {
  const int b    = blockIdx.y;
  const int wave = threadIdx.x >> 5;
  const int lane = threadIdx.x & 31;
  const int tile = blockIdx.x * 8 + wave;       // 0..111
  const int dti  = tile / 7;                    // d tile 0..15
  const int nti  = tile % 7;                    // lq tile 0..6
  const int half = lane >> 4;                   // 0/1
  const int idx  = lane & 15;
  const int dg   = dti * 16 + idx;              // M index (d)
  const int ng   = nti * 16 + idx;              // N index (lq)
  const bool nvalid = (ng < LQ_);
  const float* qrow = q + ((size_t)b * LQ_ + (nvalid ? ng : 0)) * H2_;

  v8f c0 = {}, c1 = {};
  for (int k = 0; k < H2_; k += 8) {
    const int ka = k + 2 * half;
    v2f a0, a1;
    a0.x = w[(size_t)ka * H2_ + dg];
    a0.y = w[(size_t)(ka + 1) * H2_ + dg];
    a1.x = w[(size_t)(ka + 4) * H2_ + dg];
    a1.y = w[(size_t)(ka + 5) * H2_ + dg];
    v2f b0 = *(const v2f*)(qrow + ka);          // unconditional (clamped addr)
    v2f b1 = *(const v2f*)(qrow + ka + 4);
    c0 = wmma_f32(a0, b0, c0);
    c1 = wmma_f32(a1, b1, c1);
  }
  if (nvalid) {
    for (int r = 0; r < 8; ++r) {
      const int d = dti * 16 + r + 8 * half;
      qwT[((size_t)b * H2_ + d) * LQ_ + ng] = c0[r] + c1[r];
    }
  }
}

// ---------------------------------------------------------------------------
// k2: ST[b, lp, lq] = sum_h p[b,lp,h] * qwT[b,h,lq]   (= S transposed)
//     A[m][k] = p[b,lp0+m,k] (row-major), B[k][n] = qwT[b,k,lq0+n] (row-major)
//     tiles per batch: 25 (lp) x 7 (lq) = 175 ; 8 waves/block -> 22 blocks
// ---------------------------------------------------------------------------
__global__ __launch_bounds__(256) void k2_ST(const float* __restrict__ p,
                                             const float* __restrict__ qwT,
                                             float* __restrict__ ST) {
  const int b    = blockIdx.y;
  const int wave = threadIdx.x >> 5;
  const int lane = threadIdx.x & 31;
  const int tile = blockIdx.x * 8 + wave;       // 0..175
  if (tile >= 175) return;                      // uniform per-wave exit
  const int mt   = tile / 7;                    // lp tile 0..24
  const int nt   = tile % 7;                    // lq tile 0..6
  const int half = lane >> 4;
  const int idx  = lane & 15;
  const int mg   = mt * 16 + idx;               // lp row (< 400 always)
  const int ng   = nt * 16 + idx;               // lq col
  const bool nvalid = (ng < LQ_);
  const float* prow = p + ((size_t)b * LP_ + mg) * H2_;
  const float* qcol = qwT + (size_t)b * H2_ * LQ_ + (nvalid ? ng : 0);

  v8f c0 = {}, c1 = {};
  for (int k = 0; k < H2_; k += 8) {
    const int ka = k + 2 * half;
    v2f a0 = *(const v2f*)(prow + ka);
    v2f a1 = *(const v2f*)(prow + ka + 4);
    v2f b0, b1;
    b0.x = qcol[(size_t)ka * LQ_];              // unconditional (clamped addr)
    b0.y = qcol[(size_t)(ka + 1) * LQ_];
    b1.x = qcol[(size_t)(ka + 4) * LQ_];
    b1.y = qcol[(size_t)(ka + 5) * LQ_];
    c0 = wmma_f32(a0, b0, c0);
    c1 = wmma_f32(a1, b1, c1);
  }
  if (nvalid) {
    for (int r = 0; r < 8; ++r) {
      const int m = mt * 16 + r + 8 * half;
      ST[((size_t)b * LP_ + m) * LQ_ + ng] = c0[r] + c1[r];
    }
  }
}

// ---------------------------------------------------------------------------
// k3: in-place softmax of ST rows (over LQ), emit per-row max m[b,lp].
//     one wave (32 lanes) per row; 8 rows per block; grid (50, B)
// ---------------------------------------------------------------------------
__global__ __launch_bounds__(256) void k3_softmax(float* __restrict__ ST,
                                                  float* __restrict__ mrow) {
  const int b    = blockIdx.y;
  const int wave = threadIdx.x >> 5;
  const int lane = threadIdx.x & 31;
  const int lp   = blockIdx.x * 8 + wave;       // < 400 exactly
  float* row = ST + ((size_t)b * LP_ + lp) * LQ_;

  float v[4];
  float mx = -__builtin_inff();
  for (int i = 0; i < 4; ++i) {
    const int j = lane + 32 * i;
    v[i] = (j < LQ_) ? row[j] : -__builtin_inff();
    mx = fmaxf(mx, v[i]);
  }
  for (int off = 16; off > 0; off >>= 1)
    mx = fmaxf(mx, __shfl_xor(mx, off, 32));
  float s = 0.f;
  for (int i = 0; i < 4; ++i) {
    const int j = lane + 32 * i;
    if (j < LQ_) { v[i] = __expf(v[i] - mx); s += v[i]; }
  }
  for (int off = 16; off > 0; off >>= 1)
    s += __shfl_xor(s, off, 32);
  const float inv = 1.f / s;
  for (int i = 0; i < 4; ++i) {
    const int j = lane + 32 * i;
    if (j < LQ_) row[j] = v[i] * inv;
  }
  if (lane == 0) mrow[(size_t)b * LP_ + lp] = mx;
}

// ---------------------------------------------------------------------------
// k4: bw = softmax(m over LP) ; h[b,c] = sum_lp bw[lp] * p[b,lp,c]
//     one block (256 threads) per batch
// ---------------------------------------------------------------------------
__global__ __launch_bounds__(256) void k4_h(const float* __restrict__ p,
                                            const float* __restrict__ mrow,
                                            float* __restrict__ hvec) {
  const int b = blockIdx.x;
  const int t = threadIdx.x;
  __shared__ float sred[256];
  __shared__ float bw[LP_];
  const float* m = mrow + (size_t)b * LP_;

  float mx = -__builtin_inff();
  for (int i = t; i < LP_; i += 256) mx = fmaxf(mx, m[i]);
  sred[t] = mx; __syncthreads();
  for (int off = 128; off > 0; off >>= 1) {
    if (t < off) sred[t] = fmaxf(sred[t], sred[t + off]);
    __syncthreads();
  }
  mx = sred[0]; __syncthreads();

  float s = 0.f;
  for (int i = t; i < LP_; i += 256) s += __expf(m[i] - mx);
  sred[t] = s; __syncthreads();
  for (int off = 128; off > 0; off >>= 1) {
    if (t < off) sred[t] += sred[t + off];
    __syncthreads();
  }
  const float inv = 1.f / sred[0];
  for (int i = t; i < LP_; i += 256) bw[i] = __expf(m[i] - mx) * inv;
  __syncthreads();

  float acc = 0.f;
  const float* pb = p + (size_t)b * LP_ * H2_;
  for (int lp = 0; lp < LP_; ++lp)
    acc += bw[lp] * pb[(size_t)lp * H2_ + t];
  hvec[(size_t)b * H2_ + t] = acc;
}

// ---------------------------------------------------------------------------
// k5: UT tile (16 lp x 16 ch) = a_tile(16 x 100) @ q[b](100 x 256) via WMMA,
//     then fused assembly of G = [p | h | p*U | p*h] -> out (B, LP, 1024).
//     A fragments preloaded once and reused for both channel tiles of a wave.
//     block = 8 waves; wave handles channel tiles {w, w+8}; grid (25, B)
// ---------------------------------------------------------------------------
__global__ __launch_bounds__(256) void k5_out(const float* __restrict__ p,
                                              const float* __restrict__ q,
                                              const float* __restrict__ a,  // normalized ST
                                              const float* __restrict__ hvec,
                                              float* __restrict__ out) {
  const int b    = blockIdx.y;
  const int mt   = blockIdx.x;                  // lp tile 0..24
  const int wave = threadIdx.x >> 5;
  const int lane = threadIdx.x & 31;
  const int half = lane >> 4;
  const int idx  = lane & 15;
  const int lp0  = mt * 16;
  const float* arow = a + ((size_t)b * LP_ + lp0 + idx) * LQ_;

  // Preload all A fragments for this lp-tile (K = 100 -> 25 steps of 4)
  v2f afr[25];
#pragma unroll
  for (int kk = 0; kk < 25; ++kk)
    afr[kk] = *(const v2f*)(arow + kk * 4 + 2 * half);

  const float* qb = q + (size_t)b * LQ_ * H2_;

  for (int ntp = 0; ntp < 2; ++ntp) {
    const int nt = wave + ntp * 8;              // channel tile 0..15
    const int ng = nt * 16 + idx;               // channel (< 256 always)
    v8f c0 = {}, c1 = {};
#pragma unroll
    for (int kk = 0; kk < 25; ++kk) {
      const int ka = kk * 4 + 2 * half;
      v2f bf;
      bf.x = qb[(size_t)ka * H2_ + ng];         // B[ka][n]   = q[b,ka,ng]
      bf.y = qb[(size_t)(ka + 1) * H2_ + ng];   // B[ka+1][n]
      if (kk & 1) c1 = wmma_f32(afr[kk], bf, c1);
      else        c0 = wmma_f32(afr[kk], bf, c0);
    }
    const float hv = hvec[(size_t)b * H2_ + ng];
#pragma unroll
    for (int r = 0; r < 8; ++r) {
      const int lp = lp0 + r + 8 * half;
      const float pv = p[((size_t)b * LP_ + lp) * H2_ + ng];
      float* ob = out + ((size_t)b * LP_ + lp) * (4 * H2_);
      ob[ng]           = pv;                    // P
      ob[H2_ + ng]     = hv;                    // Ht
      ob[2 * H2_ + ng] = pv * (c0[r] + c1[r]);  // P * U
      ob[3 * H2_ + ng] = pv * hv;               // P * Ht
    }
  }
}

// ---------------------------------------------------------------------------
extern "C" void kernel_launch(void* const* d_in, const int* in_sizes, int n_in,
                              void* d_out, int out_size, void* d_ws, size_t ws_size,
                              hipStream_t stream) {
  (void)in_sizes; (void)n_in; (void)out_size; (void)ws_size;
  const float* p = (const float*)d_in[0];   // (B, LP, 256)
  const float* q = (const float*)d_in[1];   // (B, LQ, 256)
  const float* w = (const float*)d_in[2];   // (256, 256)
  float* out = (float*)d_out;               // (B, LP, 1024)

  float* ws  = (float*)d_ws;
  float* qwT = ws;                                  // B*256*LQ  = 3,276,800 f
  float* ST  = qwT + (size_t)B_ * H2_ * LQ_;        // B*LP*LQ   = 5,120,000 f
  float* mr  = ST  + (size_t)B_ * LP_ * LQ_;        // B*LP      =    51,200 f
  float* hv  = mr  + (size_t)B_ * LP_;              // B*256     =    32,768 f

  k1_qwT    <<<dim3(14, B_), 256, 0, stream>>>(q, w, qwT);
  k2_ST     <<<dim3(22, B_), 256, 0, stream>>>(p, qwT, ST);
  k3_softmax<<<dim3(50, B_), 256, 0, stream>>>(ST, mr);
  k4_h      <<<dim3(B_),     256, 0, stream>>>(p, mr, hv);
  k5_out    <<<dim3(25, B_), 256, 0, stream>>>(p, q, ST, hv, out);
}